// RadialFunctionFlax_84078279787209
// MI455X (gfx1250) — compile-verified
//
#include <hip/hip_runtime.h>
#include <math.h>

// Problem constants (from the reference)
#define N_SPECIES 119
#define N_RADIAL  5
#define N_BASIS   7
#define R_MIN     0.5f
#define R_MAX     6.0f
#define PI_F      3.14159265358979323846f

typedef float v4f __attribute__((ext_vector_type(4)));
typedef int   v4i __attribute__((ext_vector_type(4)));

// betta = N_BASIS^2 / R_MAX^2 = 49/36
#define BETTA (49.0f / 36.0f)

__global__ __launch_bounds__(256) void radial_fn_kernel(
    const float* __restrict__ dr,
    const int*   __restrict__ Zi,
    const int*   __restrict__ Zj,
    const float* __restrict__ emb,   // [119][119][5][7]
    float*       __restrict__ out,   // [n][5]
    int n_edges)
{
    // rad_norm * embed_norm folded into one scale (constant-folded by compiler)
    const float rad_norm   = sqrtf(sqrtf(2.0f * BETTA / PI_F));
    const float embed_norm = rsqrtf((float)N_BASIS);
    const float norm_c     = rad_norm * embed_norm;

    const int t  = blockIdx.x * blockDim.x + threadIdx.x;
    const int e0 = t * 4;
    if (e0 >= n_edges) return;

    if (e0 + 3 < n_edges) {
        // ---- fast path: 4 edges, fully vector (b128) streamed I/O ----
        v4f d4  = __builtin_nontemporal_load((const v4f*)(dr + e0));
        v4i zi4 = __builtin_nontemporal_load((const v4i*)(Zi + e0));
        v4i zj4 = __builtin_nontemporal_load((const v4i*)(Zj + e0));

        // Kick off the L2 gather early (global_prefetch_b8) — rows are 140 B.
        int pbase[4];
#pragma unroll
        for (int i = 0; i < 4; ++i) {
            pbase[i] = (zj4[i] * N_SPECIES + zi4[i]) * (N_RADIAL * N_BASIS);
            __builtin_prefetch(emb + pbase[i], 0, 3);
            __builtin_prefetch(emb + pbase[i] + 32, 0, 3); // cover the 140B row
        }

        __attribute__((aligned(16))) float res[20];

#pragma unroll
        for (int i = 0; i < 4; ++i) {
            const float d = d4[i];

            // Gaussian radial basis (unnormalized; norm folded into `s`)
            float eb[N_BASIS];
#pragma unroll
            for (int b = 0; b < N_BASIS; ++b) {
                const float sh = R_MIN + (R_MAX - R_MIN) / (float)N_BASIS * (float)b;
                const float x  = sh - d;
                eb[b] = __expf(-BETTA * x * x);
            }

            // cosine cutoff on clipped distance
            const float dc  = fminf(d, R_MAX);
            const float cut = 0.5f * (__cosf((PI_F / R_MAX) * dc) + 1.0f);
            const float s   = cut * norm_c;

            const float* __restrict__ c = emb + pbase[i];
#pragma unroll
            for (int r = 0; r < N_RADIAL; ++r) {
                float acc = 0.0f;
#pragma unroll
                for (int b = 0; b < N_BASIS; ++b)
                    acc = fmaf(c[r * N_BASIS + b], eb[b], acc);
                res[i * N_RADIAL + r] = acc * s;
            }
        }

        // 20 contiguous, 16B-aligned floats -> 5x global_store_b128 (NT)
        v4f* o4 = (v4f*)(out + (size_t)e0 * N_RADIAL);
        const v4f* r4 = (const v4f*)res;
#pragma unroll
        for (int k = 0; k < 5; ++k)
            __builtin_nontemporal_store(r4[k], o4 + k);
    } else {
        // ---- tail path (unused when n % 4 == 0, kept for generality) ----
        for (int e = e0; e < n_edges; ++e) {
            const float d = dr[e];
            const int   p = (Zj[e] * N_SPECIES + Zi[e]) * (N_RADIAL * N_BASIS);

            float eb[N_BASIS];
#pragma unroll
            for (int b = 0; b < N_BASIS; ++b) {
                const float sh = R_MIN + (R_MAX - R_MIN) / (float)N_BASIS * (float)b;
                const float x  = sh - d;
                eb[b] = __expf(-BETTA * x * x);
            }
            const float dc  = fminf(d, R_MAX);
            const float cut = 0.5f * (__cosf((PI_F / R_MAX) * dc) + 1.0f);
            const float s   = cut * norm_c;

            const float* __restrict__ c = emb + p;
#pragma unroll
            for (int r = 0; r < N_RADIAL; ++r) {
                float acc = 0.0f;
#pragma unroll
                for (int b = 0; b < N_BASIS; ++b)
                    acc = fmaf(c[r * N_BASIS + b], eb[b], acc);
                out[(size_t)e * N_RADIAL + r] = acc * s;
            }
        }
    }
}

extern "C" void kernel_launch(void* const* d_in, const int* in_sizes, int n_in,
                              void* d_out, int out_size, void* d_ws, size_t ws_size,
                              hipStream_t stream) {
    const float* dr  = (const float*)d_in[0];
    const int*   Zi  = (const int*)  d_in[1];
    const int*   Zj  = (const int*)  d_in[2];
    const float* emb = (const float*)d_in[3];
    float*       out = (float*)d_out;

    const int n        = in_sizes[0];
    const int nThreads = (n + 3) / 4;
    const int blocks   = (nThreads + 255) / 256;

    radial_fn_kernel<<<blocks, 256, 0, stream>>>(dr, Zi, Zj, emb, out, n);
}